// ShuffleSplitBatchNormalization_51101520888529
// MI455X (gfx1250) — compile-verified
//
#include <hip/hip_runtime.h>

// ShuffleSplitBatchNorm for MI455X (gfx1250, wave32)
// x: (B=256, H=32, W=32, C=256) f32; out same. Memory bound: 768MB min traffic
// -> ~33us @ 23.3TB/s. Reduction runs on the matrix pipe (v_wmma_f32_16x16x4_f32).

#define EPSV    0.001f
#define CC      256
#define HW      1024      // 32*32
#define NB      256
#define PPART   4
#define PS      64        // batches per partition
#define NPEL    65536.0f  // PS*HW samples per (p,c)

typedef float v2f  __attribute__((ext_vector_type(2)));
typedef float v8f  __attribute__((ext_vector_type(8)));
typedef float f4v  __attribute__((ext_vector_type(4)));

// workspace layout (float units)
#define WS_PBSUM  0         // [NB*CC] per-batch channel sums
#define WS_PBSQ   65536     // [NB*CC] per-batch channel sum-of-squares
#define WS_SCALE  131072    // [PPART*CC]
#define WS_BIAS   132096    // [PPART*CC]
#define WS_PART   133120    // [NB] int partition id per batch

// ---- 1. partition assignment: part[perm[i]] = i/64 ------------------------
__global__ void k_part(const int* __restrict__ perm, int* __restrict__ part) {
    int i = threadIdx.x;           // 0..255
    part[perm[i]] = i >> 6;
}

// ---- 2. per-batch sum / sumsq via WMMA (A = ones) -------------------------
// grid = NB*2 blocks, 256 threads (8 waves). Wave w of block (b,gpair) owns
// channels [gpair*128 + w*16, +16). Each WMMA folds a 4-pixel x 16-channel
// f32 tile: D = ones(16x4) * B(4x16) + D  => every D row = per-channel sums.
__global__ __launch_bounds__(256) void k_reduce(const float* __restrict__ x,
                                                float* __restrict__ ws) {
    int b     = blockIdx.x >> 1;
    int gpair = blockIdx.x & 1;
    int t     = threadIdx.x;
    int lane  = t & 31;
    int w     = t >> 5;
    int ch    = gpair * 128 + w * 16 + (lane & 15);  // this lane's channel (N)
    int kh    = (lane >> 4) << 1;                    // pixel sub-slot: 0 or 2

    const float* base = x + (size_t)b * HW * CC + ch;

    v8f accs = {0.f, 0.f, 0.f, 0.f, 0.f, 0.f, 0.f, 0.f};
    v8f accq = {0.f, 0.f, 0.f, 0.f, 0.f, 0.f, 0.f, 0.f};
    v2f ones = {1.f, 1.f};

    for (int pp = 0; pp < HW; pp += 4) {
        float b0 = base[(pp + kh)     * CC];
        float b1 = base[(pp + kh + 1) * CC];
        v2f bv = {b0, b1};
        v2f bq = {b0 * b0, b1 * b1};
        accs = __builtin_amdgcn_wmma_f32_16x16x4_f32(false, ones, false, bv,
                                                     (short)0, accs, false, false);
        accq = __builtin_amdgcn_wmma_f32_16x16x4_f32(false, ones, false, bq,
                                                     (short)0, accq, false, false);
    }

    // D VGPR0: lanes 0-15 hold row M=0, lanes 16-31 row M=8; all rows equal
    // (A = ones), and both halves' N == lane&15 == our channel offset.
    float* pbsum = ws + WS_PBSUM;
    float* pbsq  = ws + WS_PBSQ;
    if (lane < 16) pbsum[b * CC + ch] = accs[0];
    else           pbsq [b * CC + ch] = accq[0];
}

// ---- 3. fold 64 batches per partition, pre-fold gamma/beta ----------------
__global__ __launch_bounds__(256) void k_stats(const float* __restrict__ gamma,
                                               const float* __restrict__ beta,
                                               const int* __restrict__ perm,
                                               float* __restrict__ ws) {
    int p = blockIdx.x;     // 0..3
    int c = threadIdx.x;    // 0..255
    const float* pbsum = ws + WS_PBSUM;
    const float* pbsq  = ws + WS_PBSQ;
    float s = 0.f, ss = 0.f;
    for (int i = 0; i < PS; ++i) {
        int b = perm[p * PS + i];   // batches of partition p (uniform -> SMEM)
        s  += pbsum[b * CC + c];
        ss += pbsq [b * CC + c];
    }
    float inv  = 1.0f / NPEL;
    float mean = s * inv;
    float var  = ss * inv - mean * mean;
    float sc   = gamma[p * CC + c] * rsqrtf(var + EPSV);
    float bi   = beta [p * CC + c] - mean * sc;
    ws[WS_SCALE + p * CC + c] = sc;
    ws[WS_BIAS  + p * CC + c] = bi;
}

// ---- 4. streaming normalize: 2/3 of traffic, float4 + NT hints ------------
// grid = NB*32 blocks; block = 64 channel-quads x 4 pixel slots, 8 iters
// -> 32 pixels (32KB in + 32KB out) per block, fully coalesced 16B accesses.
__global__ __launch_bounds__(256) void k_norm(const float* __restrict__ x,
                                              const float* __restrict__ ws,
                                              const int* __restrict__ part,
                                              float* __restrict__ out) {
    int blk   = blockIdx.x;
    int b     = blk >> 5;
    int chunk = blk & 31;
    int t     = threadIdx.x;
    int cq    = t & 63;     // float4 index along channels
    int psl   = t >> 6;     // pixel slot 0..3
    int p     = part[b];

    const f4v* sc4 = (const f4v*)(ws + WS_SCALE + p * CC);
    const f4v* bi4 = (const f4v*)(ws + WS_BIAS  + p * CC);
    f4v sc = sc4[cq];
    f4v bi = bi4[cq];

    const f4v* xin = (const f4v*)x;
    f4v*       o   = (f4v*)out;
    size_t base = ((size_t)b * HW + (size_t)chunk * 32) * (CC / 4);

    for (int i = 0; i < 8; ++i) {
        size_t idx = base + (size_t)(i * 4 + psl) * (CC / 4) + cq;
        f4v v = __builtin_nontemporal_load(&xin[idx]);
        v = v * sc + bi;                       // contracts to v_fma per lane
        __builtin_nontemporal_store(v, &o[idx]);
    }
}

extern "C" void kernel_launch(void* const* d_in, const int* in_sizes, int n_in,
                              void* d_out, int out_size, void* d_ws, size_t ws_size,
                              hipStream_t stream) {
    const float* x     = (const float*)d_in[0];
    const float* gamma = (const float*)d_in[1];
    const float* beta  = (const float*)d_in[2];
    const int*   perm  = (const int*)d_in[3];
    float* ws   = (float*)d_ws;
    int*   part = (int*)(ws + WS_PART);
    float* out  = (float*)d_out;

    k_part  <<<1,      256, 0, stream>>>(perm, part);
    k_reduce<<<NB * 2, 256, 0, stream>>>(x, ws);
    k_stats <<<PPART,  256, 0, stream>>>(gamma, beta, perm, ws);
    k_norm  <<<NB * 32,256, 0, stream>>>(x, ws, part, out);
}